// GATLayer_10385230922252
// MI455X (gfx1250) — compile-verified
//
#include <hip/hip_runtime.h>

typedef __attribute__((ext_vector_type(2))) float v2f;
typedef __attribute__((ext_vector_type(8))) float v8f;

#define DIN_N 128
#define DIN_E 64
#define DOUT  64
#define KCAT  (DIN_E + 2 * DOUT)   // 192
#define SLOPE 0.01f

__device__ __forceinline__ float lrelu(float x) { return x >= 0.f ? x : SLOPE * x; }

// monotone float <-> uint mapping for atomic max over signed floats
__device__ __forceinline__ unsigned fmap(float x) {
    unsigned u = __float_as_uint(x);
    return (u & 0x80000000u) ? ~u : (u | 0x80000000u);
}
__device__ __forceinline__ float funmap(unsigned u) {
    return (u & 0x80000000u) ? __uint_as_float(u ^ 0x80000000u) : __uint_as_float(~u);
}

// ---------------- init: h = 0, m_u = 0 (== -inf in mapped space), denom = 0 ----
__global__ void k_init(float* __restrict__ h, unsigned* __restrict__ m_u,
                       float* __restrict__ denom, int N) {
    int i = blockIdx.x * blockDim.x + threadIdx.x;
    if (i < N * DOUT) h[i] = 0.f;
    if (i < N) { m_u[i] = 0u; denom[i] = 0.f; }
}

// ---------------- node projection: z = nfeats @ W_fc^T  (f32 WMMA 16x16x4) ----
// one wave -> 16 nodes x 64 outputs (4 col-tiles), K = 128 in steps of 4
__global__ void k_nodeproj(const float* __restrict__ nfeats,
                           const float* __restrict__ W_fc,
                           float* __restrict__ z, int N) {
    const int lane = threadIdx.x & 31;
    const int tile = blockIdx.x * (blockDim.x >> 5) + (threadIdx.x >> 5);
    if (tile >= (N >> 4)) return;                 // wave-uniform exit
    const int r0   = tile << 4;
    const int ln   = lane & 15;
    const int koff = (lane >> 4) << 1;            // half-wave selects K pair

    const float* arow = nfeats + (size_t)(r0 + ln) * DIN_N + koff;
    const float* w0 = W_fc + (size_t)(ln)      * DIN_N + koff;
    const float* w1 = W_fc + (size_t)(16 + ln) * DIN_N + koff;
    const float* w2 = W_fc + (size_t)(32 + ln) * DIN_N + koff;
    const float* w3 = W_fc + (size_t)(48 + ln) * DIN_N + koff;

    v8f acc0 = {}, acc1 = {}, acc2 = {}, acc3 = {};
#pragma unroll 8
    for (int k = 0; k < DIN_N; k += 4) {
        v2f a  = *(const v2f*)(arow + k);
        v2f b0 = *(const v2f*)(w0 + k);
        v2f b1 = *(const v2f*)(w1 + k);
        v2f b2 = *(const v2f*)(w2 + k);
        v2f b3 = *(const v2f*)(w3 + k);
        acc0 = __builtin_amdgcn_wmma_f32_16x16x4_f32(false, a, false, b0, (short)0, acc0, false, false);
        acc1 = __builtin_amdgcn_wmma_f32_16x16x4_f32(false, a, false, b1, (short)0, acc1, false, false);
        acc2 = __builtin_amdgcn_wmma_f32_16x16x4_f32(false, a, false, b2, (short)0, acc2, false, false);
        acc3 = __builtin_amdgcn_wmma_f32_16x16x4_f32(false, a, false, b3, (short)0, acc3, false, false);
    }
    const int mbase = r0 + ((lane >> 4) << 3);    // rows split: v -> M=v / M=v+8
#pragma unroll
    for (int v = 0; v < 8; ++v) {
        size_t row = (size_t)(mbase + v) * DOUT;
        z[row +      ln] = acc0[v];
        z[row + 16 + ln] = acc1[v];
        z[row + 32 + ln] = acc2[v];
        z[row + 48 + ln] = acc3[v];
    }
}

// ---------------- edge linear + logit -----------------------------------------
// feat = lrelu([z[src] | efeat | z[dst]] @ W_edge^T + b); a = lrelu(lrelu(feat.w_coef))
// one wave -> 16 edges x 64 outputs; K = 192 in steps of 4 over three segments.
#define WMMA4(AVEC, KG)                                                                              \
    do {                                                                                             \
        v2f b0 = *(const v2f*)(w0 + (KG));                                                           \
        v2f b1 = *(const v2f*)(w1 + (KG));                                                           \
        v2f b2 = *(const v2f*)(w2 + (KG));                                                           \
        v2f b3 = *(const v2f*)(w3 + (KG));                                                           \
        acc0 = __builtin_amdgcn_wmma_f32_16x16x4_f32(false, (AVEC), false, b0, (short)0, acc0, false, false); \
        acc1 = __builtin_amdgcn_wmma_f32_16x16x4_f32(false, (AVEC), false, b1, (short)0, acc1, false, false); \
        acc2 = __builtin_amdgcn_wmma_f32_16x16x4_f32(false, (AVEC), false, b2, (short)0, acc2, false, false); \
        acc3 = __builtin_amdgcn_wmma_f32_16x16x4_f32(false, (AVEC), false, b3, (short)0, acc3, false, false); \
    } while (0)

__global__ void k_edge(const float* __restrict__ z,
                       const float* __restrict__ efeats,
                       const int* __restrict__ src,
                       const int* __restrict__ dst,
                       const float* __restrict__ W_edge,
                       const float* __restrict__ b_edge,
                       const float* __restrict__ w_coef,
                       float* __restrict__ feat,
                       float* __restrict__ a_out, int E) {
    const int lane = threadIdx.x & 31;
    const int tile = blockIdx.x * (blockDim.x >> 5) + (threadIdx.x >> 5);
    if (tile >= (E >> 4)) return;                 // wave-uniform exit
    const int e0   = tile << 4;
    const int ln   = lane & 15;
    const int koff = (lane >> 4) << 1;
    const int er   = e0 + ln;                     // this lane's gathered edge row

    const float* sbase = z      + (size_t)src[er] * DOUT  + koff;
    const float* ebase = efeats + (size_t)er      * DIN_E + koff;
    const float* dbase = z      + (size_t)dst[er] * DOUT  + koff;

    const float* w0 = W_edge + (size_t)(ln)      * KCAT + koff;
    const float* w1 = W_edge + (size_t)(16 + ln) * KCAT + koff;
    const float* w2 = W_edge + (size_t)(32 + ln) * KCAT + koff;
    const float* w3 = W_edge + (size_t)(48 + ln) * KCAT + koff;

    v8f acc0 = {}, acc1 = {}, acc2 = {}, acc3 = {};
#pragma unroll 4
    for (int k = 0; k < 64; k += 4) { v2f a = *(const v2f*)(sbase + k); WMMA4(a, k); }
#pragma unroll 4
    for (int k = 0; k < 64; k += 4) { v2f a = *(const v2f*)(ebase + k); WMMA4(a, 64 + k); }
#pragma unroll 4
    for (int k = 0; k < 64; k += 4) { v2f a = *(const v2f*)(dbase + k); WMMA4(a, 128 + k); }

    const float bi0 = b_edge[ln],      bi1 = b_edge[16 + ln];
    const float bi2 = b_edge[32 + ln], bi3 = b_edge[48 + ln];
    const float wc0 = w_coef[ln],      wc1 = w_coef[16 + ln];
    const float wc2 = w_coef[32 + ln], wc3 = w_coef[48 + ln];

    const int mbase = e0 + ((lane >> 4) << 3);
    float s[8];
#pragma unroll
    for (int v = 0; v < 8; ++v) {
        float f0 = lrelu(acc0[v] + bi0);
        float f1 = lrelu(acc1[v] + bi1);
        float f2 = lrelu(acc2[v] + bi2);
        float f3 = lrelu(acc3[v] + bi3);
        size_t row = (size_t)(mbase + v) * DOUT;
        feat[row +      ln] = f0;
        feat[row + 16 + ln] = f1;
        feat[row + 32 + ln] = f2;
        feat[row + 48 + ln] = f3;
        // partial dot for attention logit; reduce across the 16-lane half
        float p = f0 * wc0 + f1 * wc1 + f2 * wc2 + f3 * wc3;
        p += __shfl_xor(p, 1, 32);
        p += __shfl_xor(p, 2, 32);
        p += __shfl_xor(p, 4, 32);
        p += __shfl_xor(p, 8, 32);
        s[v] = p;
    }
    if (ln == 0) {
#pragma unroll
        for (int v = 0; v < 8; ++v)
            a_out[mbase + v] = lrelu(lrelu(s[v]));
    }
}

// ---------------- segment max over dst ----------------------------------------
__global__ void k_max(const float* __restrict__ a, const int* __restrict__ dst,
                      unsigned* __restrict__ m_u, int E) {
    int e = blockIdx.x * blockDim.x + threadIdx.x;
    if (e >= E) return;
    atomicMax(&m_u[dst[e]], fmap(a[e]));
}

// ---------------- exp(a - m[dst]) + segment sum -------------------------------
__global__ void k_exp(const float* __restrict__ a, const int* __restrict__ dst,
                      const unsigned* __restrict__ m_u, float* __restrict__ ex,
                      float* __restrict__ denom, int E) {
    int e = blockIdx.x * blockDim.x + threadIdx.x;
    if (e >= E) return;
    int d = dst[e];
    float v = expf(a[e] - funmap(m_u[d]));
    ex[e] = v;
    atomicAdd(&denom[d], v);
}

// ---------------- h[dst] += alpha * z[src] ------------------------------------
__global__ void k_agg(const float* __restrict__ ex, const float* __restrict__ denom,
                      const int* __restrict__ src, const int* __restrict__ dst,
                      const float* __restrict__ z, float* __restrict__ h, long long total) {
    long long gid = (long long)blockIdx.x * blockDim.x + threadIdx.x;
    if (gid >= total) return;
    int e = (int)(gid >> 6);
    int c = (int)(gid & 63);
    int d = dst[e];
    float alpha = ex[e] / denom[d];
    atomicAdd(&h[(size_t)d * DOUT + c], alpha * z[(size_t)src[e] * DOUT + c]);
}

extern "C" void kernel_launch(void* const* d_in, const int* in_sizes, int n_in,
                              void* d_out, int out_size, void* d_ws, size_t ws_size,
                              hipStream_t stream) {
    const float* nfeats = (const float*)d_in[0];
    const float* efeats = (const float*)d_in[1];
    const int*   src    = (const int*)d_in[2];
    const int*   dst    = (const int*)d_in[3];
    const float* W_fc   = (const float*)d_in[4];
    const float* W_edge = (const float*)d_in[5];
    const float* b_edge = (const float*)d_in[6];
    const float* w_coef = (const float*)d_in[7];

    const int N = in_sizes[0] / DIN_N;   // 50000 (multiple of 16)
    const int E = in_sizes[2];           // 1,600,000 (multiple of 16)

    float* h    = (float*)d_out;                         // [N*64]
    float* feat = (float*)d_out + (size_t)N * DOUT;      // [E*64]

    // workspace: z | a | ex | m_u | denom  (~26 MB)
    float*    z     = (float*)d_ws;
    float*    a     = z + (size_t)N * DOUT;
    float*    ex    = a + (size_t)E;
    unsigned* m_u   = (unsigned*)(ex + (size_t)E);
    float*    denom = (float*)(m_u + (size_t)N);

    k_init<<<(N * DOUT + 255) / 256, 256, 0, stream>>>(h, m_u, denom, N);

    int ntiles = N >> 4;                                 // 3125
    k_nodeproj<<<(ntiles + 7) / 8, 256, 0, stream>>>(nfeats, W_fc, z, N);

    int etiles = E >> 4;                                 // 100000
    k_edge<<<(etiles + 7) / 8, 256, 0, stream>>>(z, efeats, src, dst,
                                                 W_edge, b_edge, w_coef, feat, a, E);

    k_max<<<(E + 255) / 256, 256, 0, stream>>>(a, dst, m_u, E);
    k_exp<<<(E + 255) / 256, 256, 0, stream>>>(a, dst, m_u, ex, denom, E);

    long long total = (long long)E * DOUT;
    k_agg<<<(int)((total + 255) / 256), 256, 0, stream>>>(ex, denom, src, dst, z, h, total);
}